// ChamferDistanceLoss_78958678770154
// MI455X (gfx1250) — compile-verified
//
#include <hip/hip_runtime.h>

typedef float v2f __attribute__((ext_vector_type(2)));
typedef float v8f __attribute__((ext_vector_type(8)));

#define B_BATCH 8
#define NPTS 4096
#define TILES (NPTS / 16)                     // 256 column tiles
#define WAVES 8
#define THREADS (WAVES * 32)                  // 256 threads = 8 wave32
#define ROWTILES_PER_WAVE 2
#define ROWS_PER_BLOCK (WAVES * 16 * ROWTILES_PER_WAVE)   // 256 rows per block
#define BLOCKS_PER_BATCH (NPTS / ROWS_PER_BLOCK)          // 16

// layout 0: component-major [B,3,N]  (pred_pts)
// layout 1: point-major     [B,N,3]  (gt_pts)
__device__ __forceinline__ void load_point(const float* __restrict__ base, int layout,
                                           int b, int n, float& x, float& y, float& z) {
    if (layout == 0) {
        const float* p = base + (size_t)b * 3 * NPTS + n;
        x = p[0]; y = p[NPTS]; z = p[2 * NPTS];
    } else {
        const float* p = base + ((size_t)b * NPTS + n) * 3;
        x = p[0]; y = p[1]; z = p[2];
    }
}

// acc = min(acc, x, y) via one VOP3 min3 (no canonicalization traffic)
__device__ __forceinline__ void min3_acc(float& acc, float x, float y) {
    asm("v_min3_num_f32 %0, %0, %1, %2" : "+v"(acc) : "v"(x), "v"(y));
}

// For each "row" point in A, compute min over all "col" points in B of the
// squared distance, write to rowmin_out[batch*NPTS + row].
__global__ __launch_bounds__(THREADS) void chamfer_rowmin_kernel(
    const float* __restrict__ A_pts, int a_layout,
    const float* __restrict__ B_pts, int b_layout,
    float* __restrict__ rowmin_out)
{
    // 64KB LDS: per B-side point {-2x, -2y, -2z, |g|^2}
    __shared__ float4 sg[NPTS];

    const int batch  = blockIdx.x / BLOCKS_PER_BATCH;
    const int rowblk = blockIdx.x % BLOCKS_PER_BATCH;
    const int tid    = threadIdx.x;

    for (int i = tid; i < NPTS; i += THREADS) {
        float x, y, z;
        load_point(B_pts, b_layout, batch, i, x, y, z);
        float n2 = x * x + y * y + z * z;
        sg[i] = make_float4(-2.0f * x, -2.0f * y, -2.0f * z, n2);
    }
    __syncthreads();

    const int  wave   = tid >> 5;
    const int  lane   = tid & 31;
    const int  lanelo = lane & 15;
    const bool hi     = lane >= 16;
    const int  wbase  = rowblk * ROWS_PER_BLOCK + wave * (16 * ROWTILES_PER_WAVE);

    // A-matrix 16x4 f32 (ISA 7.12.2): lanes 0-15 hold K=0,1 in V0,V1;
    // lanes 16-31 hold K=2,3.  Row vector = [px, py, pz, 1].
    float ax0, ay0, az0, ax1, ay1, az1;
    load_point(A_pts, a_layout, batch, wbase + lanelo,      ax0, ay0, az0);
    load_point(A_pts, a_layout, batch, wbase + 16 + lanelo, ax1, ay1, az1);
    const float pn0 = ax0 * ax0 + ay0 * ay0 + az0 * az0;
    const float pn1 = ax1 * ax1 + ay1 * ay1 + az1 * az1;
    v2f a0, a1;
    a0.x = hi ? az0 : ax0;  a0.y = hi ? 1.0f : ay0;
    a1.x = hi ? az1 : ax1;  a1.y = hi ? 1.0f : ay1;

    float acc0[8], acc1[8];
    #pragma unroll
    for (int r = 0; r < 8; ++r) { acc0[r] = 3.0e38f; acc1[r] = 3.0e38f; }

    const float* sgf   = (const float*)sg;
    const int    laddr = (hi ? 2 : 0);

    // B-matrix 4x16 f32: lane(0..15)=N; K slot = vgpr + 2*lanehalf.
    // lanes 0-15 -> (K0,K1)=(-2gx,-2gy); lanes 16-31 -> (K2,K3)=(-2gz,gn)
    #define LOAD_BTILES(T, B0, B1)                                        \
        do {                                                              \
            const float* _p = sgf + ((size_t)(T) * 16 + lanelo) * 4 + laddr; \
            (B0).x = _p[0];  (B0).y = _p[1];                              \
            (B1).x = _p[64]; (B1).y = _p[65];                             \
        } while (0)

    #define WMMA_MIN(B0, B1)                                                                                   \
        do {                                                                                                   \
            v8f _z = {};                                                                                       \
            v8f _c00 = __builtin_amdgcn_wmma_f32_16x16x4_f32(false, a0, false, (B0), (short)0, _z, false, false); \
            v8f _c01 = __builtin_amdgcn_wmma_f32_16x16x4_f32(false, a0, false, (B1), (short)0, _z, false, false); \
            v8f _c10 = __builtin_amdgcn_wmma_f32_16x16x4_f32(false, a1, false, (B0), (short)0, _z, false, false); \
            v8f _c11 = __builtin_amdgcn_wmma_f32_16x16x4_f32(false, a1, false, (B1), (short)0, _z, false, false); \
            _Pragma("unroll")                                                                                  \
            for (int _r = 0; _r < 8; ++_r) {                                                                   \
                min3_acc(acc0[_r], _c00[_r], _c01[_r]);                                                        \
                min3_acc(acc1[_r], _c10[_r], _c11[_r]);                                                        \
            }                                                                                                  \
        } while (0)

    // Two live buffer sets (P, Q), alternated so one LDS load is always in
    // flight while the other pair feeds the matrix pipe.
    v2f pb0, pb1, qb0, qb1;
    LOAD_BTILES(0, pb0, pb1);

    for (int t = 0; t < TILES; t += 4) {
        LOAD_BTILES(t + 2, qb0, qb1);
        WMMA_MIN(pb0, pb1);
        LOAD_BTILES((t + 4) & (TILES - 1), pb0, pb1);
        WMMA_MIN(qb0, qb1);
    }
    #undef WMMA_MIN
    #undef LOAD_BTILES

    // Butterfly min across each 16-lane half (the N dimension of the tile).
    #pragma unroll
    for (int off = 1; off <= 8; off <<= 1) {
        #pragma unroll
        for (int r = 0; r < 8; ++r) {
            float o0 = __shfl_xor(acc0[r], off, 32);
            float o1 = __shfl_xor(acc1[r], off, 32);
            acc0[r] = fminf(acc0[r], o0);
            acc1[r] = fminf(acc1[r], o1);
        }
    }

    // C/D layout: lanes 0-15 hold rows 0..7 (V0..7), lanes 16-31 rows 8..15.
    // |p|^2 for row (tilebase + j) lives in lane j; gather and store from
    // lanes 0 and 16.
    const int half = hi ? 8 : 0;
    #pragma unroll
    for (int r = 0; r < 8; ++r) {
        float pn0r = __shfl(pn0, half + r, 32);
        float pn1r = __shfl(pn1, half + r, 32);
        if (lanelo == 0) {
            float* out = rowmin_out + (size_t)batch * NPTS + wbase + half + r;
            out[0]  = acc0[r] + pn0r;
            out[16] = acc1[r] + pn1r;
        }
    }
}

// loss = (sum(dist1) + sum(dist2)) / (B * N)   (valid because N == M)
__global__ __launch_bounds__(256) void chamfer_reduce_kernel(
    const float* __restrict__ d1, const float* __restrict__ d2,
    float* __restrict__ out)
{
    __shared__ float red[256];
    float s = 0.0f;
    const int n = B_BATCH * NPTS;
    for (int i = threadIdx.x; i < n; i += 256) s += d1[i] + d2[i];
    red[threadIdx.x] = s;
    __syncthreads();
    for (int off = 128; off > 0; off >>= 1) {
        if (threadIdx.x < off) red[threadIdx.x] += red[threadIdx.x + off];
        __syncthreads();
    }
    if (threadIdx.x == 0) out[0] = red[0] / (float)(B_BATCH * NPTS);
}

extern "C" void kernel_launch(void* const* d_in, const int* in_sizes, int n_in,
                              void* d_out, int out_size, void* d_ws, size_t ws_size,
                              hipStream_t stream) {
    const float* pred = (const float*)d_in[0];  // [8, 3, 4096] component-major
    const float* gt   = (const float*)d_in[1];  // [8, 4096, 3] point-major

    float* dist1 = (float*)d_ws;                 // [8*4096]
    float* dist2 = dist1 + (size_t)B_BATCH * NPTS;

    dim3 grid(B_BATCH * BLOCKS_PER_BATCH);       // 128 blocks
    // Pass 1: rows = pred points, cols = gt points  -> dist1
    chamfer_rowmin_kernel<<<grid, THREADS, 0, stream>>>(pred, 0, gt, 1, dist1);
    // Pass 2: rows = gt points, cols = pred points  -> dist2
    chamfer_rowmin_kernel<<<grid, THREADS, 0, stream>>>(gt, 1, pred, 0, dist2);

    chamfer_reduce_kernel<<<1, 256, 0, stream>>>(dist1, dist2, (float*)d_out);
}